// RGATEncoder_52647709114670
// MI455X (gfx1250) — compile-verified
//
#include <hip/hip_runtime.h>

// RGAT encoder for MI455X (gfx1250): fp32 WMMA projections with async-LDS
// double-buffered staging + scalar-logit attention with L2-resident atomics.

#define NN 50000
#define MM 8
#define EE 800000
#define RR 8
#define DD 128
#define HH 128
#define GG 64
#define SLOPE 0.2f

typedef __attribute__((ext_vector_type(2))) float v2f;
typedef __attribute__((ext_vector_type(8))) float v8f;
typedef int v4i_vs __attribute__((vector_size(16)));   // matches builtin param

#ifndef __has_builtin
#define __has_builtin(x) 0
#endif

#if __has_builtin(__builtin_amdgcn_global_load_async_to_lds_b128)
#define HAVE_ASYNC_LDS 1
#else
#define HAVE_ASYNC_LDS 0
#endif

// 16B global -> LDS copy; async (ASYNCcnt-tracked DMA) when available.
__device__ __forceinline__ void cp16(const float* __restrict__ g, float* l) {
#if HAVE_ASYNC_LDS
  __builtin_amdgcn_global_load_async_to_lds_b128(
      (__attribute__((address_space(1))) v4i_vs*)g,
      (__attribute__((address_space(3))) v4i_vs*)l, 0, 0);
#else
  *(float4*)l = *(const float4*)g;
#endif
}

__device__ __forceinline__ void async_wait0() {
#if HAVE_ASYNC_LDS
#if __has_builtin(__builtin_amdgcn_s_wait_asynccnt)
  __builtin_amdgcn_s_wait_asynccnt(0);
#else
  asm volatile("s_wait_asynccnt 0x0" ::: "memory");
#endif
#endif
}

__device__ __forceinline__ void atomAddF(float* p, float v) {
  __hip_atomic_fetch_add(p, v, __ATOMIC_RELAXED, __HIP_MEMORY_SCOPE_AGENT);
}

// Monotonic float<->uint mapping so atomicMax(uint) == float max (order-free,
// deterministic). enc is strictly increasing in f.
__device__ __forceinline__ unsigned enc_f32(float f) {
  unsigned u = __float_as_uint(f);
  return (u & 0x80000000u) ? ~u : (u | 0x80000000u);
}
__device__ __forceinline__ float dec_f32(unsigned e) {
  return (e & 0x80000000u) ? __uint_as_float(e & 0x7fffffffu)
                           : __uint_as_float(~e);
}

__global__ __launch_bounds__(256) void k_zero(float* __restrict__ p, long n) {
  long i = (long)blockIdx.x * blockDim.x + threadIdx.x;
  long stride = (long)gridDim.x * blockDim.x;
  for (; i < n; i += stride) p[i] = 0.0f;
}

// One wave (32 lanes) per node; each lane owns 4 feature dims (float4).
__global__ __launch_bounds__(256) void k_embed_mean(
    const int* __restrict__ xpad, const float* __restrict__ emb,
    float* __restrict__ x) {
  int node = blockIdx.x * 8 + (threadIdx.x >> 5);
  if (node >= NN) return;
  int lane = threadIdx.x & 31;
  float4 acc = {0.f, 0.f, 0.f, 0.f};
  float cnt = 0.f;
  for (int j = 0; j < MM; ++j) {
    int idx = xpad[node * MM + j];           // uniform across the wave
    if (idx >= 0) {
      float4 v = *(const float4*)(emb + (size_t)idx * DD + lane * 4);
      acc.x += v.x; acc.y += v.y; acc.z += v.z; acc.w += v.w;
      cnt += 1.f;
    }
  }
  float inv = 1.f / cnt;                     // lengths >= 1 guaranteed
  acc.x *= inv; acc.y *= inv; acc.z *= inv; acc.w *= inv;
  *(float4*)(x + (size_t)node * DD + lane * 4) = acc;
}

// wq[r] = w[r] @ q, wk[r] = w[r] @ k  (tiny: 8x128 dots of length 128)
__global__ __launch_bounds__(256) void k_wqk(
    const float* __restrict__ w, const float* __restrict__ q,
    const float* __restrict__ k, float* __restrict__ wq,
    float* __restrict__ wk) {
  int id = blockIdx.x * 256 + threadIdx.x;   // (r,f) flattened, 0..1023
  if (id >= RR * DD) return;
  const float* wrow = w + (size_t)id * HH;
  float sq = 0.f, sk = 0.f;
  for (int o = 0; o < HH; ++o) {
    float wv = wrow[o];
    sq += wv * q[o];
    sk += wv * k[o];
  }
  wq[id] = sq;
  wk[id] = sk;
}

// a_i[n,r] = dot(x[n], wq[r]) ; a_j[n,r] = dot(x[n], wk[r]). Wave per node.
__global__ __launch_bounds__(256) void k_node_terms(
    const float* __restrict__ x, const float* __restrict__ wq,
    const float* __restrict__ wk, float* __restrict__ ai,
    float* __restrict__ aj) {
  int node = blockIdx.x * 8 + (threadIdx.x >> 5);
  if (node >= NN) return;
  int lane = threadIdx.x & 31;
  float4 xv = *(const float4*)(x + (size_t)node * DD + lane * 4);
  for (int r = 0; r < RR; ++r) {
    float4 qv = *(const float4*)(wq + r * DD + lane * 4);
    float4 kv = *(const float4*)(wk + r * DD + lane * 4);
    float sq = xv.x * qv.x + xv.y * qv.y + xv.z * qv.z + xv.w * qv.w;
    float sk = xv.x * kv.x + xv.y * kv.y + xv.z * kv.z + xv.w * kv.w;
    for (int off = 16; off > 0; off >>= 1) {
      sq += __shfl_xor(sq, off, 32);
      sk += __shfl_xor(sk, off, 32);
    }
    if (lane == 0) {
      ai[node * RR + r] = sq;
      aj[node * RR + r] = sk;
    }
  }
}

// WMMA projection: xw[n, r, :] = x[n, :] @ w[r, :, :]   (fp32 16x16x4 WMMA)
// Block = 128 threads (4 waves); tile = 64 nodes x 128 outputs. W streamed
// through LDS in double-buffered K-chunks of 16 via async-to-LDS DMA so the
// waves keep issuing WMMA while the next chunk is in flight. Rows padded to
// 132 floats (stride % 64 = 4 banks) to kill A-column bank conflicts.
#define KC 16
__global__ __launch_bounds__(128) void k_proj_wmma(
    const float* __restrict__ X, const float* __restrict__ W,
    float* __restrict__ XW) {
  __shared__ float lA[64][132];
  __shared__ float lW[2][KC][132];
  const int r = blockIdx.y;
  const int node0 = blockIdx.x * 64;
  const int tid = threadIdx.x;
  const float* Wr = W + (size_t)r * DD * HH;

  // Issue A tile (64x128, zero-padded past N) + W chunk 0.
  for (int i = tid; i < 64 * 32; i += 128) {   // 32 float4 per row
    int row = i >> 5, col4 = i & 31;
    int node = node0 + row;
    if (node < NN)
      cp16(X + (size_t)node * DD + col4 * 4, &lA[row][col4 * 4]);
    else {
      float4 z = {0.f, 0.f, 0.f, 0.f};
      *(float4*)(&lA[row][col4 * 4]) = z;
    }
  }
  for (int i = tid; i < KC * 32; i += 128) {
    int row = i >> 5, col4 = i & 31;
    cp16(Wr + (size_t)row * HH + col4 * 4, &lW[0][row][col4 * 4]);
  }
  async_wait0();
  __syncthreads();

  const int lane = tid & 31;
  const int wave = tid >> 5;                   // 0..3 -> rows wave*16..+15
  const int half = lane >> 4;                  // 0/1 (K or M/N half select)
  const int q = lane & 15;

  v8f acc[8];
#pragma unroll
  for (int c = 0; c < 8; ++c) {
    v8f z = {0.f, 0.f, 0.f, 0.f, 0.f, 0.f, 0.f, 0.f};
    acc[c] = z;
  }

  const int T = DD / KC;                       // 8 chunks
  for (int t = 0; t < T; ++t) {
    int cur = t & 1;
    if (t + 1 < T) {                           // prefetch next chunk (async)
      int kc2 = (t + 1) * KC;
      for (int i = tid; i < KC * 32; i += 128) {
        int row = i >> 5, col4 = i & 31;
        cp16(Wr + (size_t)(kc2 + row) * HH + col4 * 4,
             &lW[cur ^ 1][row][col4 * 4]);
      }
    }
    int kc = t * KC;
#pragma unroll
    for (int k0 = 0; k0 < KC; k0 += 4) {
      int ka = k0 + 2 * half;
      v2f a = *(const v2f*)(&lA[wave * 16 + q][kc + ka]);
#pragma unroll
      for (int c = 0; c < 8; ++c) {
        v2f b;
        b.x = lW[cur][ka][c * 16 + q];
        b.y = lW[cur][ka + 1][c * 16 + q];
        acc[c] = __builtin_amdgcn_wmma_f32_16x16x4_f32(
            false, a, false, b, (short)0, acc[c], false, false);
      }
    }
    if (t + 1 < T) {
      async_wait0();                           // next chunk landed
      __syncthreads();                         // visible; readers of cur done
    }
  }

#pragma unroll
  for (int c = 0; c < 8; ++c) {
#pragma unroll
    for (int v = 0; v < 8; ++v) {
      int node = node0 + wave * 16 + v + 8 * half;
      if (node < NN)
        XW[((size_t)node * RR + r) * HH + c * 16 + q] = acc[c][v];
    }
  }
}

// Pass 1 over edges: logit = leaky_relu(a_i[dst,t] + a_j[src,t]); running max
// per dst via monotonic-uint atomicMax.
__global__ __launch_bounds__(256) void k_edge_logit(
    const int* __restrict__ ei, const int* __restrict__ et,
    const float* __restrict__ ai, const float* __restrict__ aj,
    float* __restrict__ logits, unsigned* __restrict__ mu) {
  int e = blockIdx.x * 256 + threadIdx.x;
  if (e >= EE) return;
  int s = ei[e], d = ei[EE + e], t = et[e];
  float l = ai[d * RR + t] + aj[s * RR + t];
  l = l > 0.f ? l : SLOPE * l;
  logits[e] = l;
  atomicMax(mu + d, enc_f32(l));
}

// Pass 2: y[dst] += exp(l - m[dst]) * xw[src,t,:], denom[dst] += exp(...).
// One wave per edge; lane owns 4 channels. Softmax normalization is folded
// into the per-node epilogue (y/denom), saving a whole edge pass.
__global__ __launch_bounds__(256) void k_edge_accum(
    const int* __restrict__ ei, const int* __restrict__ et,
    const float* __restrict__ logits, const unsigned* __restrict__ mu,
    const float* __restrict__ xw, float* __restrict__ denom,
    float* __restrict__ y) {
  int e = blockIdx.x * 8 + (threadIdx.x >> 5);
  if (e >= EE) return;
  int lane = threadIdx.x & 31;
  int s = ei[e], d = ei[EE + e], t = et[e];
  float ev = __expf(logits[e] - dec_f32(mu[d]));
  if (lane == 0) atomAddF(denom + d, ev);
  float4 xj = *(const float4*)(xw + ((size_t)s * RR + t) * HH + lane * 4);
  float* yd = y + (size_t)d * HH + lane * 4;
  atomAddF(yd + 0, ev * xj.x);
  atomAddF(yd + 1, ev * xj.y);
  atomAddF(yd + 2, ev * xj.z);
  atomAddF(yd + 3, ev * xj.w);
}

// x_out[n] = relu(y[n]/denom[n] + bias); empty dst (denom==0) -> relu(bias).
__global__ __launch_bounds__(256) void k_epilogue(
    const float* __restrict__ y, const float* __restrict__ denom,
    const float* __restrict__ b, float* __restrict__ xout) {
  int node = blockIdx.x * 8 + (threadIdx.x >> 5);
  if (node >= NN) return;
  int lane = threadIdx.x & 31;
  float dnm = denom[node];
  float inv = dnm > 0.f ? 1.f / dnm : 0.f;
  float4 v = *(const float4*)(y + (size_t)node * HH + lane * 4);
  float4 bb = *(const float4*)(b + lane * 4);
  float4 o;
  o.x = fmaxf(v.x * inv + bb.x, 0.f);
  o.y = fmaxf(v.y * inv + bb.y, 0.f);
  o.z = fmaxf(v.z * inv + bb.z, 0.f);
  o.w = fmaxf(v.w * inv + bb.w, 0.f);
  *(float4*)(xout + (size_t)node * HH + lane * 4) = o;
}

// Final: z[n] = x[n] @ lin_w^T + lin_b, pooled into out[batch[n]] (WMMA +
// atomicAdd; out is only 64x128 so the atomics stay in L2).
__global__ __launch_bounds__(128) void k_final_wmma(
    const float* __restrict__ X, const float* __restrict__ linw,
    const float* __restrict__ linb, const int* __restrict__ batch,
    float* __restrict__ out) {
  __shared__ float lA[64][132];
  __shared__ float lB[32][132];
  const int node0 = blockIdx.x * 64;
  const int tid = threadIdx.x;

  for (int i = tid; i < 64 * 32; i += 128) {
    int row = i >> 5, col4 = i & 31;
    int node = node0 + row;
    if (node < NN)
      cp16(X + (size_t)node * DD + col4 * 4, &lA[row][col4 * 4]);
    else {
      float4 z = {0.f, 0.f, 0.f, 0.f};
      *(float4*)(&lA[row][col4 * 4]) = z;
    }
  }

  const int lane = tid & 31;
  const int wave = tid >> 5;
  const int half = lane >> 4;
  const int q = lane & 15;

  v8f acc[8];
#pragma unroll
  for (int c = 0; c < 8; ++c) {
    v8f z = {0.f, 0.f, 0.f, 0.f, 0.f, 0.f, 0.f, 0.f};
    acc[c] = z;
  }

  for (int kc = 0; kc < HH; kc += 32) {
    // B[k][o] = lin_w[o][kc+k] (transpose while staging; scatter -> no async)
    for (int i = tid; i < 32 * 128; i += 128) {
      int h = i & 31, o = i >> 5;
      lB[h][o] = linw[(size_t)o * HH + kc + h];
    }
    async_wait0();                             // A tile (first iter only)
    __syncthreads();
#pragma unroll
    for (int k0 = 0; k0 < 32; k0 += 4) {
      int ka = k0 + 2 * half;
      v2f a = *(const v2f*)(&lA[wave * 16 + q][kc + ka]);
#pragma unroll
      for (int c = 0; c < 8; ++c) {
        v2f b;
        b.x = lB[ka][c * 16 + q];
        b.y = lB[ka + 1][c * 16 + q];
        acc[c] = __builtin_amdgcn_wmma_f32_16x16x4_f32(
            false, a, false, b, (short)0, acc[c], false, false);
      }
    }
    __syncthreads();                           // readers done before restage
  }

  int gidx[8];
#pragma unroll
  for (int v = 0; v < 8; ++v) {
    int node = node0 + wave * 16 + v + 8 * half;
    gidx[v] = (node < NN) ? batch[node] : -1;
  }
#pragma unroll
  for (int c = 0; c < 8; ++c) {
    int n = c * 16 + q;
    float bn = linb[n];
#pragma unroll
    for (int v = 0; v < 8; ++v) {
      if (gidx[v] >= 0)
        atomAddF(out + (size_t)gidx[v] * HH + n, acc[c][v] + bn);
    }
  }
}

extern "C" void kernel_launch(void* const* d_in, const int* in_sizes, int n_in,
                              void* d_out, int out_size, void* d_ws,
                              size_t ws_size, hipStream_t stream) {
  (void)in_sizes; (void)n_in; (void)out_size; (void)ws_size;
  const int* xpad = (const int*)d_in[0];
  const int* ei = (const int*)d_in[1];
  const int* et = (const int*)d_in[2];
  const int* batch = (const int*)d_in[3];
  const float* emb = (const float*)d_in[4];
  const float* w1 = (const float*)d_in[5];
  const float* q1 = (const float*)d_in[6];
  const float* k1 = (const float*)d_in[7];
  const float* b1 = (const float*)d_in[8];
  const float* w2 = (const float*)d_in[9];
  const float* q2 = (const float*)d_in[10];
  const float* k2 = (const float*)d_in[11];
  const float* b2 = (const float*)d_in[12];
  const float* linw = (const float*)d_in[13];
  const float* linb = (const float*)d_in[14];
  float* out = (float*)d_out;

  float* ws = (float*)d_ws;
  float* x0 = ws;      ws += (size_t)NN * DD;
  float* x1 = ws;      ws += (size_t)NN * DD;
  float* xw = ws;      ws += (size_t)NN * RR * HH;
  float* ai = ws;      ws += (size_t)NN * RR;
  float* aj = ws;      ws += (size_t)NN * RR;
  float* wq = ws;      ws += RR * DD;
  float* wk = ws;      ws += RR * DD;
  float* logits = ws;  ws += EE;
  unsigned* mu = (unsigned*)ws; ws += NN;       // contiguous: mu, denom, y
  float* denom = ws;   ws += NN;
  float* y = ws;       ws += (size_t)NN * HH;

  const dim3 b256(256);
  const int nodeBlocks = (NN + 7) / 8;          // 8 waves (=8 nodes) / block

  k_embed_mean<<<nodeBlocks, b256, 0, stream>>>(xpad, emb, x0);

  const float* Wl[2] = {w1, w2};
  const float* Ql[2] = {q1, q2};
  const float* Kl[2] = {k1, k2};
  const float* Bl[2] = {b1, b2};
  float* xin = x0;
  float* xout = x1;
  for (int layer = 0; layer < 2; ++layer) {
    k_wqk<<<4, b256, 0, stream>>>(Wl[layer], Ql[layer], Kl[layer], wq, wk);
    k_node_terms<<<nodeBlocks, b256, 0, stream>>>(xin, wq, wk, ai, aj);
    dim3 gproj((NN + 63) / 64, RR);
    k_proj_wmma<<<gproj, dim3(128), 0, stream>>>(xin, Wl[layer], xw);
    k_zero<<<1024, b256, 0, stream>>>((float*)mu, (long)NN * (2 + HH));
    k_edge_logit<<<(EE + 255) / 256, b256, 0, stream>>>(ei, et, ai, aj,
                                                        logits, mu);
    k_edge_accum<<<(EE + 7) / 8, b256, 0, stream>>>(ei, et, logits, mu, xw,
                                                    denom, y);
    k_epilogue<<<nodeBlocks, b256, 0, stream>>>(y, denom, Bl[layer], xout);
    float* t = xin; xin = xout; xout = t;
  }

  k_zero<<<32, b256, 0, stream>>>(out, (long)GG * HH);
  k_final_wmma<<<(NN + 63) / 64, dim3(128), 0, stream>>>(xin, linw, linb,
                                                         batch, out);
}